// SATNet_49349174231090
// MI455X (gfx1250) — compile-verified
//
#include <hip/hip_runtime.h>
#include <hip/hip_bf16.h>
#include <math.h>

typedef __attribute__((ext_vector_type(2))) float v2f;
typedef __attribute__((ext_vector_type(8))) float v8f;
typedef __attribute__((address_space(3))) float lds_f3;

#define Bq    32
#define NVq   400
#define Nq    501
#define Mq    512
#define Kq    32
#define NPAD  504   // N padded to multiple of 4 for the WMMA K-loop
#define VS    33    // padded V row stride (33 coprime with 64 banks)

__device__ __forceinline__ float wsum(float v) {
#pragma unroll
  for (int off = 16; off > 0; off >>= 1) v += __shfl_xor(v, off, 32);
  return v;
}

// Async global->LDS copy (CDNA5): 8 bytes/lane, tracked by ASYNCcnt.
__device__ __forceinline__ void async_row_b64(float* dst_generic, const float* src) {
  unsigned lds_off = (unsigned)(uintptr_t)(lds_f3*)dst_generic;
  asm volatile("global_load_async_to_lds_b64 %0, %1, off"
               :: "v"(lds_off), "v"(src) : "memory");
}

__device__ __forceinline__ void wait_async0() {
  asm volatile("s_wait_asynccnt 0x0" ::: "memory");
}

__global__ __launch_bounds__(256) void satnet_mix_kernel(
    const float* __restrict__ S, const float* __restrict__ z,
    const int* __restrict__ is_input, const float* __restrict__ Vinit,
    const int* __restrict__ perm, float* __restrict__ out)
{
  __shared__ float Vl[NPAD * VS];                 // ~65 KB : V (padded stride)
  __shared__ float Wl[Kq * Mq];                   //  64 KB : W[k][m]
  __shared__ __align__(16) float sbuf2[2][Mq];    //  4 KB  : double-buffered S row
  __shared__ float Snl[NPAD];                     //  row norms^2 of S
  __shared__ float zfl[NPAD];                     //  z_full
  __shared__ int   inpl[NPAD];                    //  pin mask
  __shared__ float gbuf[Kq];
  __shared__ float dvbuf[Kq];

  const int b    = blockIdx.x;      // one batch per workgroup
  const int tid  = threadIdx.x;     // 256 threads = 8 waves (wave32)
  const int lane = tid & 31;
  const int w    = tid >> 5;
  const bool lo  = lane < 16;
  const int r    = lane & 15;

  // ---------------- z_full / inp ----------------
  for (int n = tid; n < NPAD; n += 256) {
    float zv; int ip;
    if (n == 0)        { zv = 1.f; ip = 1; }
    else if (n <= NVq) { zv = z[b * NVq + n - 1]; ip = (is_input[b * NVq + n - 1] != 0); }
    else               { zv = 0.f; ip = 0; }
    zfl[n] = zv; inpl[n] = ip;
  }

  // ---------------- Snrms (wave per row) ----------------
  for (int n = w; n < Nq; n += 8) {
    const float* sr = S + n * Mq;
    float p = 0.f;
#pragma unroll
    for (int t = 0; t < 16; ++t) { float x = sr[lane + (t << 5)]; p = fmaf(x, x, p); }
    p = wsum(p);
    if (lane == 0) Snl[n] = p;
  }

  // ---------------- normalize V rows (pad rows -> 0) ----------------
  for (int n = w; n < NPAD; n += 8) {
    float v = (n < Nq) ? Vinit[(b * Nq + n) * Kq + lane] : 0.f;
    float nn = sqrtf(wsum(v * v));
    Vl[n * VS + lane] = v / fmaxf(nn, 1e-12f);
  }
  __syncthreads();

  // ---------------- Vperp + pinning ----------------
  for (int n = w; n < Nq; n += 8) {
    if (n == 0) continue;                       // row 0 stays = v0
    float v0v = Vl[lane];                       // row 0
    float v   = Vl[n * VS + lane];
    float d   = wsum(v * v0v);
    float vp  = v - d * v0v;
    float pn  = sqrtf(wsum(vp * vp));
    vp = vp / fmaxf(pn, 1e-12f);
    float ang = 3.14159265358979323846f * zfl[n];
    float pinned = -cosf(ang) * v0v + sinf(ang) * vp;
    if (inpl[n]) Vl[n * VS + lane] = pinned;    // wave-uniform branch
  }
  __syncthreads();

  // ---------------- W = V^T * S  via V_WMMA_F32_16X16X4_F32 ----------------
  // D (32 x 512) per batch: 2 row-tiles (k) x 32 col-tiles (m); 8 waves x 4 col-tiles.
  {
    v8f zz = {0.f, 0.f, 0.f, 0.f, 0.f, 0.f, 0.f, 0.f};
    v8f acc[4][2];
#pragma unroll
    for (int q = 0; q < 4; ++q) { acc[q][0] = zz; acc[q][1] = zz; }

    const int nA = lo ? 0 : 2;                  // A/B K-slot per lane half
    for (int nb = 0; nb < NPAD / 4; ++nb) {
      const int n0 = nb * 4;
      const int ra = n0 + nA;                   // K row for .x
      const int rb = ra + 1;                    // K row for .y
      // A = V^T : A[m=k][K=n] = V[n][k]; rows >= Nq are zero in Vl
      v2f a0, a1;
      a0.x = Vl[ra * VS + r];       a0.y = Vl[rb * VS + r];
      a1.x = Vl[ra * VS + 16 + r];  a1.y = Vl[rb * VS + 16 + r];
#pragma unroll
      for (int q = 0; q < 4; ++q) {
        const int m = ((w << 2) + q) * 16 + r;
        v2f bf;
        bf.x = (ra < Nq) ? S[ra * Mq + m] : 0.f;
        bf.y = (rb < Nq) ? S[rb * Mq + m] : 0.f;
        acc[q][0] = __builtin_amdgcn_wmma_f32_16x16x4_f32(
            false, a0, false, bf, (short)0, acc[q][0], false, false);
        acc[q][1] = __builtin_amdgcn_wmma_f32_16x16x4_f32(
            false, a1, false, bf, (short)0, acc[q][1], false, false);
      }
    }
    // store D tiles into LDS W (C/D layout: vgpr j -> row j (+8 for hi lanes))
#pragma unroll
    for (int q = 0; q < 4; ++q) {
      const int m = ((w << 2) + q) * 16 + r;
#pragma unroll
      for (int kt = 0; kt < 2; ++kt)
#pragma unroll
        for (int j = 0; j < 8; ++j) {
          const int row = kt * 16 + (lo ? j : j + 8);
          Wl[row * Mq + m] = acc[q][kt][j];
        }
    }
  }

  // ---------------- 10 sweeps of 500 sequential coordinate steps ----------------
  const int m0 = tid << 1;
  // prologue: async-prefetch row perm[0] into buffer 0
  {
    const int i0 = perm[0];
    async_row_b64(&sbuf2[0][m0], S + i0 * Mq + m0);
  }
  int p = 0;
  for (int it = 0; it < 10; ++it) {
    for (int jj = 0; jj < Nq - 1; ++jj) {
      const int i = perm[jj];                   // uniform -> scalar load
      wait_async0();                            // row i resident in sbuf2[p]
      __syncthreads();                          // + previous step's W update done

      // async-prefetch next step's row into the other buffer (overlaps the step)
      const int jn  = (jj + 1 < Nq - 1) ? jj + 1 : 0;
      const int inx = perm[jn];
      async_row_b64(&sbuf2[p ^ 1][m0], S + inx * Mq + m0);

      const float* sb = sbuf2[p];

      // g[k] = W[k,:]·s - Snrms[i]*v_old[k] ; wave w owns k = 4w..4w+3
      const float sn = Snl[i];
#pragma unroll
      for (int kk = 0; kk < 4; ++kk) {
        const int k = (w << 2) + kk;
        const float* wr = &Wl[k * Mq];
        float pa = 0.f;
#pragma unroll
        for (int t = 0; t < 16; ++t) { const int m = lane + (t << 5); pa = fmaf(wr[m], sb[m], pa); }
        pa = wsum(pa);
        if (lane == 0) gbuf[k] = pa - sn * Vl[i * VS + k];
      }
      __syncthreads();

      // wave 0: normalize, pin, compute delta, update V row i
      if (w == 0) {
        float g  = gbuf[lane];
        float n2 = wsum(g * g);
        float vo = Vl[i * VS + lane];
        float vn = -g / fmaxf(sqrtf(n2), 1e-12f);
        if (inpl[i]) vn = vo;
        dvbuf[lane] = vn - vo;
        Vl[i * VS + lane] = vn;
      }
      __syncthreads();

      // rank-1 update: W[k][m] += dv[k]*s[m]; each thread owns 2 m-columns
      const float s0 = sb[m0], s1 = sb[m0 + 1];
#pragma unroll 8
      for (int k = 0; k < Kq; ++k) {
        const float d = dvbuf[k];
        Wl[k * Mq + m0]     = fmaf(d, s0, Wl[k * Mq + m0]);
        Wl[k * Mq + m0 + 1] = fmaf(d, s1, Wl[k * Mq + m0 + 1]);
      }
      // no trailing barrier: next step's wait+barrier orders everything
      p ^= 1;
    }
  }
  __syncthreads();

  // ---------------- output: z_out = acos(clip(-V·v0))/pi ----------------
  for (int idx = w; idx < NVq; idx += 8) {
    const int n = idx + 1;
    float d = wsum(Vl[n * VS + lane] * Vl[lane]);   // V[n]·v0
    if (lane == 0) {
      float c  = fminf(fmaxf(-d, -1.f + 1e-7f), 1.f - 1e-7f);
      float zv = acosf(c) * 0.31830988618379067154f;   // 1/pi
      if (inpl[n]) zv = zfl[n];
      out[b * NVq + idx] = zv;
    }
  }
}

extern "C" void kernel_launch(void* const* d_in, const int* in_sizes, int n_in,
                              void* d_out, int out_size, void* d_ws, size_t ws_size,
                              hipStream_t stream) {
  (void)in_sizes; (void)n_in; (void)out_size; (void)d_ws; (void)ws_size;
  const float* S        = (const float*)d_in[0];
  const float* z        = (const float*)d_in[1];
  const int*   is_input = (const int*)d_in[2];
  const float* Vinit    = (const float*)d_in[3];
  const int*   perm     = (const int*)d_in[4];
  float* out = (float*)d_out;
  satnet_mix_kernel<<<dim3(Bq), dim3(256), 0, stream>>>(S, z, is_input, Vinit, perm, out);
}